// RoIPoolingLayer_29910152249561
// MI455X (gfx1250) — compile-verified
//
#include <hip/hip_runtime.h>
#include <hip/hip_bf16.h>

// Problem constants (fixed by the reference setup).
#define B_   4
#define H_   56
#define W_   56
#define C_   256
#define R_   64
#define OH_  7
#define OW_  7
#define WSEG 31   // max ROI width from setup (len < 0.55 -> w1-w0 <= 31)

typedef int v4i __attribute__((ext_vector_type(4)));
typedef __attribute__((address_space(1))) v4i* gv4i_p;   // global int4*
typedef __attribute__((address_space(3))) v4i* lv4i_p;   // LDS int4*

// ---- CDNA5 async global->LDS copy (128 bits per lane) -----------------------
__device__ __forceinline__ void async_copy_b128(const void* gsrc, void* ldst) {
#if __has_builtin(__builtin_amdgcn_global_load_async_to_lds_b128)
    __builtin_amdgcn_global_load_async_to_lds_b128(
        (gv4i_p)gsrc, (lv4i_p)ldst, /*imm offset=*/0, /*cpol=*/0);
#else
    unsigned lds_off = (unsigned)(size_t)ldst;
    unsigned long long gaddr = (unsigned long long)(size_t)gsrc;
    asm volatile("global_load_async_to_lds_b128 %0, %1, off"
                 :
                 : "v"(lds_off), "v"(gaddr)
                 : "memory");
#endif
}

__device__ __forceinline__ void wait_async_zero() {
#if __has_builtin(__builtin_amdgcn_s_wait_asynccnt)
    __builtin_amdgcn_s_wait_asynccnt(0);
#else
    asm volatile("s_wait_asynccnt 0x0" ::: "memory");
#endif
}

// ---- Bin arithmetic, mirroring the reference exactly ------------------------
// step = int32(float32(end-start) / nbins); bin i = [start+i*step, start+(i+1)*step),
// last bin extends to end; step==0 => all valid pixels land in the last bin.
__device__ __forceinline__ void bin_range(int s, int e, int nb, int i,
                                          int& lo, int& hi) {
    int step = (int)((float)(e - s) / (float)nb);
    if (step > 0) {
        lo = s + i * step;
        hi = (i == nb - 1) ? e : (s + (i + 1) * step);
    } else {
        lo = (i == nb - 1) ? s : 0;
        hi = (i == nb - 1) ? e : 0;
    }
}

__global__ __launch_bounds__(C_) void roi_maxpool_kernel(
    const float* __restrict__ fmap,   // [B, H, W, C]
    const float* __restrict__ rois,   // [B, R, 4]
    float* __restrict__ out)          // [B, R, OH, OW, C]
{
    // Double-buffered row segments: 2 * 31 * 256 * 4 = 62 KB LDS.
    __shared__ __align__(16) float smem[2][WSEG * C_];

    const int tid = threadIdx.x;                 // channel index
    const int blk = blockIdx.x;                  // (b*R + r)*OH + oh
    const int oh  = blk % OH_;
    const int br  = blk / OH_;
    const int r   = br % R_;
    const int b   = br / R_;

    const float* roi = rois + (size_t)(b * R_ + r) * 4;
    const int h0 = (int)((float)H_ * roi[0]);
    const int w0 = (int)((float)W_ * roi[1]);
    const int h1 = (int)((float)H_ * roi[2]);
    const int w1 = (int)((float)W_ * roi[3]);

    int rs, re;
    bin_range(h0, h1, OH_, oh, rs, re);

    int width = w1 - w0;
    if (width < 0) width = 0;

    // Column bin ranges relative to w0 (registers, fully unrolled use below).
    int xs[OW_], xe[OW_];
    {
        const int stepw = (int)((float)(w1 - w0) / (float)OW_);
#pragma unroll
        for (int ow = 0; ow < OW_; ++ow) {
            if (stepw > 0) {
                xs[ow] = ow * stepw;
                xe[ow] = (ow == OW_ - 1) ? width : (ow + 1) * stepw;
            } else {
                xs[ow] = 0;
                xe[ow] = (ow == OW_ - 1) ? width : 0;
            }
        }
    }

    const float NEG_INF = -__builtin_huge_valf();
    float acc[OW_];
#pragma unroll
    for (int ow = 0; ow < OW_; ++ow) acc[ow] = NEG_INF;

    // Base of the ROI's row segments; one row segment is contiguous in memory.
    const char* gbase =
        (const char*)(fmap + (size_t)((b * H_) * W_ + w0) * C_);
    const size_t row_bytes = (size_t)W_ * C_ * 4;   // stride between rows

    // Generic path tiles the segment in WSEG-column chunks; with the harness
    // inputs (width <= 31) this loop executes exactly once.
    for (int x0 = 0; x0 < width; x0 += WSEG) {
        const int wseg = (width - x0 < WSEG) ? (width - x0) : WSEG;
        const int seg_chunks = wseg * (C_ * 4 / 16);    // 16-byte chunks
        const size_t seg_off = (size_t)x0 * C_ * 4;

        __syncthreads();   // buffers idle before (re)priming the pipeline

        // Prime: issue async DMA for the first row of this bin.
        if (rs < re) {
            const char* gsrc = gbase + (size_t)rs * row_bytes + seg_off;
            char* lbase = (char*)smem[0];
            for (int i = tid; i < seg_chunks; i += C_)
                async_copy_b128(gsrc + (size_t)i * 16, lbase + (size_t)i * 16);
        }

        for (int h = rs; h < re; ++h) {
            const int cur = (h - rs) & 1;

            wait_async_zero();   // row h (and any older) DMA complete (this wave)
            __syncthreads();     // all waves' row-h data visible in LDS

            // Overlap: kick off row h+1 DMA into the other buffer while we
            // consume row h. Buffer (cur^1) was last read at iteration h-1,
            // strictly before the barrier above -> safe to overwrite.
            if (h + 1 < re) {
                const char* gsrc = gbase + (size_t)(h + 1) * row_bytes + seg_off;
                char* lbase = (char*)smem[cur ^ 1];
                for (int i = tid; i < seg_chunks; i += C_)
                    async_copy_b128(gsrc + (size_t)i * 16,
                                    lbase + (size_t)i * 16);
            }

            // Per-channel max into 7 register accumulators. LDS reads are
            // bank-conflict-free: lane t -> word x*256 + t (64 banks, wave32).
#pragma unroll
            for (int ow = 0; ow < OW_; ++ow) {
                int lo = xs[ow] > x0 ? xs[ow] : x0;
                int hi = xe[ow] < x0 + wseg ? xe[ow] : x0 + wseg;
                float a = acc[ow];
                for (int x = lo - x0; x < hi - x0; ++x) {
                    a = fmaxf(a, smem[cur][x * C_ + tid]);
                }
                acc[ow] = a;
            }
        }
    }

    float* o = out + (size_t)((b * R_ + r) * OH_ + oh) * (OW_ * C_) + tid;
#pragma unroll
    for (int ow = 0; ow < OW_; ++ow) {
        o[ow * C_] = acc[ow];
    }
}

extern "C" void kernel_launch(void* const* d_in, const int* in_sizes, int n_in,
                              void* d_out, int out_size, void* d_ws, size_t ws_size,
                              hipStream_t stream) {
    (void)in_sizes; (void)n_in; (void)out_size; (void)d_ws; (void)ws_size;
    const float* fmap = (const float*)d_in[0];  // [4,56,56,256] fp32
    const float* rois = (const float*)d_in[1];  // [4,64,4]      fp32
    float* out = (float*)d_out;                 // [4,64,7,7,256] fp32

    dim3 grid(B_ * R_ * OH_);
    dim3 block(C_);
    roi_maxpool_kernel<<<grid, block, 0, stream>>>(fmap, rois, out);
}